// KnowledgeModule_8194797601190
// MI455X (gfx1250) — compile-verified
//
#include <hip/hip_runtime.h>
#include <math.h>

// ---------------------------------------------------------------------------
// Sparse knowledge-module forward (4 gather/segment-reduce layers).
// Memory-bound: ~60-70MB DRAM traffic -> ~3us at 23.3TB/s. No GEMM -> no WMMA.
// CDNA5 paths used: double-buffered async global->LDS gather (ASYNCcnt with
// partial waits), global_prefetch_b8, non-temporal index-stream loads.
// ---------------------------------------------------------------------------

#define BLK 256
#define OUT0 2097152
#define OUT1 262144
#define OUT2 65536
#define OUT3 16384

typedef int v4i __attribute__((ext_vector_type(4)));

#if defined(__gfx1250__) &&                                            \
    __has_builtin(__builtin_amdgcn_global_load_async_to_lds_b32) &&    \
    __has_builtin(__builtin_amdgcn_s_wait_asynccnt)
#define USE_ASYNC_GATHER 1
// Builtin expects typed pointers: (global int*, lds int*, imm offset, imm cpol)
typedef __attribute__((address_space(1))) int* glob_i32p;
typedef __attribute__((address_space(3))) int* lds_i32p;
#define ASYNC_GATHER_B32(gsrc, ldst)                                   \
  __builtin_amdgcn_global_load_async_to_lds_b32(                       \
      (glob_i32p)(void*)(gsrc), (lds_i32p)(void*)(ldst), 0, 0)
#else
#define USE_ASYNC_GATHER 0
#endif

__device__ __forceinline__ float dev_log1mexp(float x) {
  // x is strictly negative (log-probability). Matches reference log1mexp.
  const float cutoff = -0.6931471805599453f; // -ln(2)
  return (x > cutoff) ? logf(-expm1f(x)) : log1pf(-expf(x));
}

// Decode one edge value from its index + (possibly bogus) gathered x_pos value.
// x[0] = -inf, x[1] = 0, x[2+2i] = x_pos[i], x[3+2i] = log1mexp(x_pos[i])
__device__ __forceinline__ float decode_edge(unsigned u, float g) {
  if (u == 0u) return -__builtin_inff();              // log(0)
  if (u == 1u) return 0.0f;                           // log(1)
  return (u & 1u) ? dev_log1mexp(g) : g;
}

// ---- Layer 0: fused encode + gather + fan-4 segment sum -------------------
__global__ void k_layer0(const float* __restrict__ xp,
                         const int* __restrict__ ix,
                         float* __restrict__ y, int n_out) {
  const int stride = blockDim.x * gridDim.x;
  int o = blockIdx.x * blockDim.x + threadIdx.x;

#if USE_ASYNC_GATHER
  // Double-buffered per-lane async gather: while decoding group i from LDS
  // buffer b, group i+1 is in flight into buffer b^1. ASYNCcnt partial wait
  // (<=4) drains only the older group (async loads complete in order).
  __shared__ float tile[2][BLK * 4];

  bool valid = o < n_out;           // wave-uniform: sizes are multiples of stride
  v4i e_cur = {};
  if (valid) {
    e_cur = __builtin_nontemporal_load((const v4i*)ix + o);
#pragma unroll
    for (int j = 0; j < 4; ++j) {
      unsigned u = (unsigned)e_cur[j];
      unsigned var = (u >= 2u) ? ((u - 2u) >> 1) : 0u;  // units clamped; overridden in decode
      ASYNC_GATHER_B32(xp + var, &tile[0][threadIdx.x + BLK * j]);
    }
  }

  int buf = 0;
  while (valid) {
    const int o_next = o + stride;
    const bool next_valid = o_next < n_out;
    v4i e_next = e_cur;
    if (next_valid) {
      // Run-ahead prefetch of the index quad two iterations out (streaming).
      if (o_next + stride < n_out)
        __builtin_prefetch(&ix[4 * (size_t)(o_next + stride)], 0, 0);
      e_next = __builtin_nontemporal_load((const v4i*)ix + o_next);
#pragma unroll
      for (int j = 0; j < 4; ++j) {
        unsigned u = (unsigned)e_next[j];
        unsigned var = (u >= 2u) ? ((u - 2u) >> 1) : 0u;
        ASYNC_GATHER_B32(xp + var, &tile[buf ^ 1][threadIdx.x + BLK * j]);
      }
      __builtin_amdgcn_s_wait_asynccnt(4);   // drain only the older group
    } else {
      __builtin_amdgcn_s_wait_asynccnt(0);   // epilogue: drain everything
    }
    asm volatile("" ::: "memory");

    float acc = 0.0f;
#pragma unroll
    for (int j = 0; j < 4; ++j)
      acc += decode_edge((unsigned)e_cur[j], tile[buf][threadIdx.x + BLK * j]);
    y[o] = acc;

    o = o_next;
    e_cur = e_next;
    buf ^= 1;
    valid = next_valid;
  }
#else
  for (; o < n_out; o += stride) {
    v4i e = __builtin_nontemporal_load((const v4i*)ix + o);
    float acc = 0.0f;
#pragma unroll
    for (int j = 0; j < 4; ++j) {
      unsigned u = (unsigned)e[j];
      unsigned var = (u >= 2u) ? ((u - 2u) >> 1) : 0u;
      acc += decode_edge(u, xp[var]);
    }
    y[o] = acc;
  }
#endif
}

// ---- LogSumExp layer (fan = FAN), reference-exact semantics ---------------
template <int FAN>
__global__ void k_lse(const float* __restrict__ xin,
                      const int* __restrict__ ix,
                      float* __restrict__ y, int n_out) {
  int o = blockIdx.x * blockDim.x + threadIdx.x;
  if (o >= n_out) return;
  float v[FAN];
#pragma unroll
  for (int q = 0; q < FAN / 4; ++q) {
    v4i e = __builtin_nontemporal_load((const v4i*)ix + (o * (FAN / 4) + q));
    v[4 * q + 0] = xin[e.x];
    v[4 * q + 1] = xin[e.y];
    v[4 * q + 2] = xin[e.z];
    v[4 * q + 3] = xin[e.w];
  }
  float m = v[0];
#pragma unroll
  for (int j = 1; j < FAN; ++j) m = fmaxf(m, v[j]);
  float s = 0.0f;
#pragma unroll
  for (int j = 0; j < FAN; ++j) {
    float d = v[j] - m;          // NaN only when v == m == -inf
    d = (d != d) ? 0.0f : d;     // nan_to_num_(0)
    s += expf(d);
  }
  y[o] = logf(s + 1e-15f) + m;   // m == -inf propagates -inf correctly
}

// ---- Fan-4 segment sum (layer 2) ------------------------------------------
__global__ void k_sum4(const float* __restrict__ xin,
                       const int* __restrict__ ix,
                       float* __restrict__ y, int n_out) {
  int o = blockIdx.x * blockDim.x + threadIdx.x;
  if (o >= n_out) return;
  v4i e = __builtin_nontemporal_load((const v4i*)ix + o);
  y[o] = xin[e.x] + xin[e.y] + xin[e.z] + xin[e.w];
}

// ---------------------------------------------------------------------------
extern "C" void kernel_launch(void* const* d_in, const int* in_sizes, int n_in,
                              void* d_out, int out_size, void* d_ws, size_t ws_size,
                              hipStream_t stream) {
  (void)in_sizes; (void)n_in; (void)out_size; (void)ws_size;
  // Input order: x_pos, ix_in0, seg0, ix_in1, seg1, ix_in2, seg2, ix_in3, seg3
  // seg arrays are repeat(arange(n_out), fan) -> contiguous segments -> unused.
  const float* x_pos = (const float*)d_in[0];
  const int*   ix0   = (const int*)d_in[1];
  const int*   ix1   = (const int*)d_in[3];
  const int*   ix2   = (const int*)d_in[5];
  const int*   ix3   = (const int*)d_in[7];

  float* y0  = (float*)d_ws;     // 2M floats  (8 MB)
  float* y1  = y0 + OUT0;        // 256K floats (1 MB)
  float* y2  = y1 + OUT1;        // 64K floats (0.25 MB)
  float* out = (float*)d_out;    // 16K floats

  // Layer 0: grid-stride (4 iterations/thread) so the async pipeline and the
  // index-stream prefetch have depth; 2048 blocks x 256 threads saturates.
  k_layer0<<<2048, BLK, 0, stream>>>(x_pos, ix0, y0, OUT0);
  k_lse<8><<<OUT1 / BLK, BLK, 0, stream>>>(y0, ix1, y1, OUT1);
  k_sum4 <<<OUT2 / BLK, BLK, 0, stream>>>(y1, ix2, y2, OUT2);
  k_lse<4><<<OUT3 / BLK, BLK, 0, stream>>>(y2, ix3, out, OUT3);
}